// EuclideanEmbedding_11802570129617
// MI455X (gfx1250) — compile-verified
//
#include <hip/hip_runtime.h>
#include <hip/hip_bf16.h>
#include <math.h>

#define BATCH 8192
#define NVEC  512
#define DIM   64

typedef float v2f __attribute__((ext_vector_type(2)));
typedef float v8f __attribute__((ext_vector_type(8)));

// ---------------------------------------------------------------------------
// Kernel 1: per-row squared norms for x (8192 rows) and p (512 rows), plus
// initialization of the min arrays to +inf. One wave32 per row.
// ---------------------------------------------------------------------------
__global__ __launch_bounds__(256) void eucl_norms_init_kernel(
    const float* __restrict__ x, const float* __restrict__ p,
    float* __restrict__ xx, float* __restrict__ pp,
    unsigned* __restrict__ rowmin, unsigned* __restrict__ colmin)
{
    const int lane = threadIdx.x & 31;
    const int wave = threadIdx.x >> 5;
    const int row  = blockIdx.x * 8 + wave;
    if (row >= BATCH + NVEC) return;

    const float* src = (row < BATCH) ? (x + (size_t)row * DIM)
                                     : (p + (size_t)(row - BATCH) * DIM);
    v2f v = *(const v2f*)(src + lane * 2);
    float s = v.x * v.x + v.y * v.y;
    #pragma unroll
    for (int off = 16; off >= 1; off >>= 1)
        s += __shfl_xor(s, off, 32);

    if (lane == 0) {
        if (row < BATCH) {
            xx[row] = s;
            rowmin[row] = 0x7f800000u;           // +inf
        } else {
            pp[row - BATCH] = s;
            colmin[row - BATCH] = 0x7f800000u;   // +inf
        }
    }
}

// ---------------------------------------------------------------------------
// Kernel 2: WMMA cdist. One wave per 16x16 output tile; K=64 via 16 chained
// v_wmma_f32_16x16x4_f32. Fused min-reductions via uint atomicMin (distances
// are non-negative so float bits compare monotonically as unsigned).
//
// fp32 A(16x4) fragment layout (ISA 7.12.2): lane L holds row M=L%16,
//   VGPR0 -> K = 2*(L>=16), VGPR1 -> K+1.  B(4x16) is the mirror image with
//   N=L%16, so both operands are a single float2 load per K-step.
// C/D layout: VGPR r, lane L -> M = r + 8*(L>=16), N = L%16.
// ---------------------------------------------------------------------------
__global__ __launch_bounds__(256) void eucl_cdist_wmma_kernel(
    const float* __restrict__ x, const float* __restrict__ p,
    const float* __restrict__ xx, const float* __restrict__ pp,
    float* __restrict__ dist,
    unsigned* __restrict__ rowmin, unsigned* __restrict__ colmin)
{
    const int lane  = threadIdx.x & 31;
    const int wave  = threadIdx.x >> 5;
    const int tile  = blockIdx.x * 8 + wave;   // 16384 tiles total
    const int mTile = tile >> 5;               // 512 row tiles
    const int nTile = tile & 31;               // 32  col tiles
    const int mBase = mTile * 16;
    const int nBase = nTile * 16;
    const int l16   = lane & 15;
    const int koff  = (lane >> 4) << 1;        // 0 or 2

    const float* arow = x + (size_t)(mBase + l16) * DIM + koff;
    const float* brow = p + (size_t)(nBase + l16) * DIM + koff;

    v8f c = {0.f, 0.f, 0.f, 0.f, 0.f, 0.f, 0.f, 0.f};
    #pragma unroll
    for (int kk = 0; kk < DIM / 4; ++kk) {
        v2f a = *(const v2f*)(arow + kk * 4);
        v2f b = *(const v2f*)(brow + kk * 4);
        // D = A x B + C ; exact fp32 matrix op (16x16x4)
        c = __builtin_amdgcn_wmma_f32_16x16x4_f32(
                /*neg_a=*/false, a, /*neg_b=*/false, b,
                /*c_mod=*/(short)0, c, /*reuse_a=*/false, /*reuse_b=*/false);
    }

    const int   half8 = (lane >> 4) << 3;      // 0 or 8
    const float ppn   = pp[nBase + l16];
    float cmin = __builtin_inff();

    #pragma unroll
    for (int r = 0; r < 8; ++r) {
        const int   m  = mBase + r + half8;
        const float d2 = xx[m] + ppn - 2.0f * c[r];
        const float dd = sqrtf(fmaxf(d2, 0.0f));
        dist[(size_t)m * NVEC + nBase + l16] = dd;
        cmin = fminf(cmin, dd);

        // min over the 16 columns of this row (shuffles stay in each half)
        float rmin = dd;
        #pragma unroll
        for (int off = 8; off >= 1; off >>= 1)
            rmin = fminf(rmin, __shfl_xor(rmin, off, 32));
        if (l16 == 0)
            atomicMin(&rowmin[m], __float_as_uint(rmin));
    }

    // min over the 16 rows of this column: fold upper half onto lower half
    cmin = fminf(cmin, __shfl_xor(cmin, 16, 32));
    if (lane < 16)
        atomicMin(&colmin[nBase + l16], __float_as_uint(cmin));
}

// ---------------------------------------------------------------------------
// Kernel 3: means of the min arrays -> r1 (colmin/512), r2 (rowmin/8192).
// ---------------------------------------------------------------------------
__global__ __launch_bounds__(256) void eucl_finalize_kernel(
    const unsigned* __restrict__ rowmin, const unsigned* __restrict__ colmin,
    float* __restrict__ out)
{
    __shared__ float s_r[256];
    __shared__ float s_c[256];
    const int tid = threadIdx.x;

    float sr = 0.f, sc = 0.f;
    for (int i = tid; i < BATCH; i += 256) sr += __uint_as_float(rowmin[i]);
    for (int i = tid; i < NVEC;  i += 256) sc += __uint_as_float(colmin[i]);
    s_r[tid] = sr; s_c[tid] = sc;
    __syncthreads();
    for (int off = 128; off >= 1; off >>= 1) {
        if (tid < off) { s_r[tid] += s_r[tid + off]; s_c[tid] += s_c[tid + off]; }
        __syncthreads();
    }
    if (tid == 0) {
        out[(size_t)BATCH * NVEC]     = s_c[0] / (float)NVEC;   // r1_cost
        out[(size_t)BATCH * NVEC + 1] = s_r[0] / (float)BATCH;  // r2_cost
    }
}

extern "C" void kernel_launch(void* const* d_in, const int* in_sizes, int n_in,
                              void* d_out, int out_size, void* d_ws, size_t ws_size,
                              hipStream_t stream) {
    const float* x = (const float*)d_in[0];   // (8192, 64) fp32
    const float* p = (const float*)d_in[1];   // (512, 64)  fp32
    float* out = (float*)d_out;               // distances flat + r1 + r2

    // workspace layout (69,632 bytes)
    float*    xx     = (float*)d_ws;              // 8192
    float*    pp     = xx + BATCH;                // 512
    unsigned* rowmin = (unsigned*)(pp + NVEC);    // 8192
    unsigned* colmin = rowmin + BATCH;            // 512

    // 8704 rows, 8 waves/block
    eucl_norms_init_kernel<<<(BATCH + NVEC) / 8, 256, 0, stream>>>(
        x, p, xx, pp, rowmin, colmin);

    // 512*32 = 16384 tiles, 8 waves/block -> 2048 blocks
    eucl_cdist_wmma_kernel<<<(BATCH / 16) * (NVEC / 16) / 8, 256, 0, stream>>>(
        x, p, xx, pp, out, rowmin, colmin);

    eucl_finalize_kernel<<<1, 256, 0, stream>>>(rowmin, colmin, out);
}